// VariationalGCNEncoder_979252543686
// MI455X (gfx1250) — compile-verified
//
#include <hip/hip_runtime.h>
#include <hip/hip_bf16.h>

typedef __attribute__((ext_vector_type(2))) float v2f;
typedef __attribute__((ext_vector_type(8))) float v8f;

#define H_DIM 128

// ---------------- degree / norm ----------------
__global__ void k_init_deg(float* deg, int N) {
    int i = blockIdx.x * blockDim.x + threadIdx.x;
    if (i < N) deg[i] = 1.0f;   // self-loop weight
}

__global__ void k_accum_deg(const int* __restrict__ cols, const float* __restrict__ w,
                            float* deg, int E) {
    int e = blockIdx.x * blockDim.x + threadIdx.x;
    if (e < E) unsafeAtomicAdd(&deg[cols[e]], w[e]);
}

__global__ void k_dinv(float* deg, int N) {
    int i = blockIdx.x * blockDim.x + threadIdx.x;
    if (i < N) {
        float d = deg[i];
        deg[i] = d > 0.0f ? rsqrtf(d) : 0.0f;
    }
}

// ---------------- WMMA f32 GEMM: Y[N x 128] = op(X)[N x 128] @ W[128 x 128] ----------------
template <bool RELU>
__device__ inline v2f load_a_frag(const float* __restrict__ X, int row0, int k0, int lane) {
    int m  = lane & 15;
    int kk = (lane & 16) ? (k0 + 2) : k0;     // lanes 0-15: K=k0,k0+1 ; lanes 16-31: K=k0+2,k0+3
    const float* p = X + (size_t)(row0 + m) * H_DIM + kk;
    v2f a;
    a.x = p[0];
    a.y = p[1];
    if (RELU) { a.x = fmaxf(a.x, 0.0f); a.y = fmaxf(a.y, 0.0f); }
    return a;
}

__device__ inline v2f load_b_frag(const float* __restrict__ W, int k0, int col0, int lane) {
    int n  = lane & 15;
    int kk = (lane & 16) ? (k0 + 2) : k0;
    v2f b;
    b.x = W[(size_t)(kk + 0) * H_DIM + col0 + n];
    b.y = W[(size_t)(kk + 1) * H_DIM + col0 + n];
    return b;
}

template <bool RELU>
__global__ void k_gemm128(const float* __restrict__ X, const float* __restrict__ W,
                          float* __restrict__ Y) {
    // block = 64 threads = 2 waves; wave w computes rows [blockIdx.x*16, +16) x cols [w*64, +64)
    int lane = threadIdx.x & 31;
    int wave = threadIdx.x >> 5;
    int row0 = blockIdx.x * 16;
    int col0 = wave * 64;

    v8f c0 = {}, c1 = {}, c2 = {}, c3 = {};
#pragma unroll
    for (int k0 = 0; k0 < H_DIM; k0 += 4) {
        v2f a  = load_a_frag<RELU>(X, row0, k0, lane);
        v2f b0 = load_b_frag(W, k0, col0 +  0, lane);
        v2f b1 = load_b_frag(W, k0, col0 + 16, lane);
        v2f b2 = load_b_frag(W, k0, col0 + 32, lane);
        v2f b3 = load_b_frag(W, k0, col0 + 48, lane);
        c0 = __builtin_amdgcn_wmma_f32_16x16x4_f32(false, a, false, b0, (short)0, c0, false, false);
        c1 = __builtin_amdgcn_wmma_f32_16x16x4_f32(false, a, false, b1, (short)0, c1, false, false);
        c2 = __builtin_amdgcn_wmma_f32_16x16x4_f32(false, a, false, b2, (short)0, c2, false, false);
        c3 = __builtin_amdgcn_wmma_f32_16x16x4_f32(false, a, false, b3, (short)0, c3, false, false);
    }

    int m_off = (lane & 16) ? 8 : 0;   // C/D: vgpr r -> rows r (lanes 0-15) and r+8 (lanes 16-31)
    int n     = lane & 15;
#pragma unroll
    for (int r = 0; r < 8; ++r) {
        float* yp = Y + (size_t)(row0 + r + m_off) * H_DIM + n + col0;
        yp[ 0] = c0[r];
        yp[16] = c1[r];
        yp[32] = c2[r];
        yp[48] = c3[r];
    }
}

// ---------------- fused (bias + self-loop) init ----------------
// dst(i,f) = bias(f) + dinv[i]^2 * H[i,f], with f<64 -> dstLo (+f), f>=64 -> dstHi (+f-64)
__global__ void k_init_agg(const float* __restrict__ Hm, const float* __restrict__ dinv,
                           const float* __restrict__ biasLo, const float* __restrict__ biasHi,
                           float* __restrict__ dstLo, float* __restrict__ dstHi,
                           int ld, int N) {
    int tid = blockIdx.x * blockDim.x + threadIdx.x;
    int i = tid >> 5;
    if (i >= N) return;
    int lane = tid & 31;
    int f = lane * 4;
    float s = dinv[i];
    s = s * s;
    float4 hv = *(const float4*)(Hm + (size_t)i * H_DIM + f);
    const float* bp = (f < 64) ? (biasLo + f) : (biasHi + (f - 64));
    float* dp = (f < 64) ? (dstLo + (size_t)i * ld + f) : (dstHi + (size_t)i * ld + (f - 64));
    dp[0] = bp[0] + s * hv.x;
    dp[1] = bp[1] + s * hv.y;
    dp[2] = bp[2] + s * hv.z;
    dp[3] = bp[3] + s * hv.w;
}

// ---------------- edge SpMM: dst[col] += dinv[row]*w*dinv[col] * H[row] ----------------
__global__ void k_spmm(const int* __restrict__ rows, const int* __restrict__ cols,
                       const float* __restrict__ w, const float* __restrict__ dinv,
                       const float* __restrict__ Hm,
                       float* dstLo, float* dstHi, int ld, int E) {
    int tid = blockIdx.x * blockDim.x + threadIdx.x;
    int e = tid >> 5;                 // one wave32 per edge
    if (e >= E) return;
    int lane = tid & 31;
    int r = rows[e];
    int c = cols[e];
    float nw = dinv[r] * w[e] * dinv[c];
    int f = lane * 4;
    float4 hv = *(const float4*)(Hm + (size_t)r * H_DIM + f);
    float* dp = (f < 64) ? (dstLo + (size_t)c * ld + f) : (dstHi + (size_t)c * ld + (f - 64));
    unsafeAtomicAdd(dp + 0, nw * hv.x);
    unsafeAtomicAdd(dp + 1, nw * hv.y);
    unsafeAtomicAdd(dp + 2, nw * hv.z);
    unsafeAtomicAdd(dp + 3, nw * hv.w);
}

// ---------------- Wcat = [Wmu | Wls] (128 x 128) ----------------
__global__ void k_build_wcat(const float* __restrict__ Wmu, const float* __restrict__ Wls,
                             float* __restrict__ Wcat) {
    int idx = blockIdx.x * blockDim.x + threadIdx.x;   // 16384 threads
    int k = idx >> 7;
    int j = idx & 127;
    Wcat[idx] = (j < 64) ? Wmu[k * 64 + j] : Wls[k * 64 + (j - 64)];
}

extern "C" void kernel_launch(void* const* d_in, const int* in_sizes, int n_in,
                              void* d_out, int out_size, void* d_ws, size_t ws_size,
                              hipStream_t stream) {
    const int*   e_row = (const int*)d_in[0];            // edge_index[0]  (E)
    const float* e_w   = (const float*)d_in[1];
    const float* emb   = (const float*)d_in[2];
    const float* W1    = (const float*)d_in[3];
    const float* b1    = (const float*)d_in[4];
    const float* Wmu   = (const float*)d_in[5];
    const float* bmu   = (const float*)d_in[6];
    const float* Wls   = (const float*)d_in[7];
    const float* bls   = (const float*)d_in[8];

    const int E = in_sizes[1];
    const int N = in_sizes[2] / H_DIM;                   // 50000 (divisible by 16)
    const int* e_col = e_row + E;                        // edge_index[1]

    float* out = (float*)d_out;                          // mu [N*64] then logstd [N*64]

    float* ws    = (float*)d_ws;
    float* dinv  = ws;                                   // N          (deg -> dinv in place)
    float* h1    = dinv + N;                             // N*128      (also reused as t)
    float* agg1  = h1 + (size_t)N * H_DIM;               // N*128
    float* wcat  = agg1 + (size_t)N * H_DIM;             // 128*128

    const int TB = 256;

    // 1) normalization coefficients
    k_init_deg<<<(N + TB - 1) / TB, TB, 0, stream>>>(dinv, N);
    k_accum_deg<<<(E + TB - 1) / TB, TB, 0, stream>>>(e_col, e_w, dinv, E);
    k_dinv<<<(N + TB - 1) / TB, TB, 0, stream>>>(dinv, N);

    // 2) h1 = emb @ W1
    k_gemm128<false><<<N / 16, 64, 0, stream>>>(emb, W1, h1);

    // 3) agg1 = b1 + dinv^2*h1 ; agg1[col] += norm * h1[row]
    long long nodeWaves = (long long)N * 32;
    long long edgeWaves = (long long)E * 32;
    k_init_agg<<<(int)((nodeWaves + TB - 1) / TB), TB, 0, stream>>>(
        h1, dinv, b1, b1 + 64, agg1, agg1 + 64, H_DIM, N);
    k_spmm<<<(int)((edgeWaves + TB - 1) / TB), TB, 0, stream>>>(
        e_row, e_col, e_w, dinv, h1, agg1, agg1 + 64, H_DIM, E);

    // 4) t = relu(agg1) @ [Wmu|Wls]   (t aliases h1, which is dead now)
    k_build_wcat<<<(H_DIM * H_DIM) / TB, TB, 0, stream>>>(Wmu, Wls, wcat);
    k_gemm128<true><<<N / 16, 64, 0, stream>>>(agg1, wcat, h1);

    // 5) out = bias + dinv^2*t ; out[col] += norm * t[row]  (mu | logstd split)
    k_init_agg<<<(int)((nodeWaves + TB - 1) / TB), TB, 0, stream>>>(
        h1, dinv, bmu, bls, out, out + (size_t)N * 64, 64, N);
    k_spmm<<<(int)((edgeWaves + TB - 1) / TB), TB, 0, stream>>>(
        e_row, e_col, e_w, dinv, h1, out, out + (size_t)N * 64, 64, E);
}